// DAO_87909390615208
// MI455X (gfx1250) — compile-verified
//
#include <hip/hip_runtime.h>
#include <math.h>

// Problem constants (match reference)
#define NB 8
#define HH 64
#define WW 64
#define C_ 192
#define G_ 12
#define GC_ 16
#define P_ 9
#define L_ (NB*HH*WW)          // 32768 pixels
#define GP2_ (G_*P_*2)         // 216
#define GP_ (G_*P_)            // 108

typedef float v2f __attribute__((ext_vector_type(2)));
typedef float v8f __attribute__((ext_vector_type(8)));

// -------------------------------------------------------------------------
// Weight pack: Bt[n][k] = (n < N) ? B[k][n] : 0   for n in [0, Npad)
// Transposing makes each lane's B-fragment a contiguous (k,k+1) pair
// (-> global_load_b64), and zero-padding removes all bounds checks from
// the GEMM K-loop. Weights are tiny (<= 216x192), cost is negligible.
// -------------------------------------------------------------------------
__global__ __launch_bounds__(256) void pack_bt(
    const float* __restrict__ B, float* __restrict__ Bt,
    int N, int K, int total /* = Npad*K */)
{
  int tid = blockIdx.x * 256 + threadIdx.x;
  if (tid >= total) return;
  int n = tid / K;
  int k = tid - n * K;
  Bt[n * K + k] = (n < N) ? B[k * N + n] : 0.0f;
}

// -------------------------------------------------------------------------
// fp32 WMMA GEMM:  C[M,N] = A[M,K] @ B[K,N] + bias[N]
// B pre-packed transposed+padded as Bt[Npad][K]. One wave computes a
// 32(M) x 64(N) tile: 8 accumulators; per k-step 2 A-fragments and 4
// B-fragments feed 8 WMMAs (A reused 4x, B reused 2x -> 8 WMMA : 6 loads).
// K-loop is branch-free: b64 loads + pointer increments only.
//
// ISA layouts (CDNA5 ISA §7.12.2):
//   A 16x4 f32 : lane<16 -> M=lane, {v0,v1}={K0,K1}; lane>=16 -> {K2,K3}
//   B  4x16 f32: lane<16 -> N=lane, {v0,v1}=rows{K0,K1}; lane>=16 rows{K2,K3}
//   C/D 16x16  : VGPR r: lane<16 -> (M=r, N=lane); lane>=16 -> (M=r+8)
// -------------------------------------------------------------------------
__global__ __launch_bounds__(128) void gemm_wmma_f32(
    const float* __restrict__ A, const float* __restrict__ Bt,
    const float* __restrict__ bias, float* __restrict__ Cout,
    int N, int K, int tilesN4, int totalWaves)
{
  int wave = blockIdx.x * (blockDim.x >> 5) + (threadIdx.x >> 5);
  if (wave >= totalWaves) return;               // wave-uniform exit
  int tm = wave / tilesN4;
  int tn = wave - tm * tilesN4;
  int m0 = tm << 5;        // 32-row tile
  int n0 = tn << 6;        // 64-col strip

  int lane  = threadIdx.x & 31;
  int laneN = lane & 15;
  int hi    = lane >> 4;   // half-wave select
  int kb    = hi << 1;     // K sub-offset {0,2}

  // All offsets even floats -> 8-byte aligned v2f loads.
  const float* ap = A  + (m0 + laneN) * K + kb;
  const float* bp = Bt + (n0 + laneN) * K + kb;

  v8f acc[8] = {{}, {}, {}, {}, {}, {}, {}, {}};

  #pragma unroll 2
  for (int k0 = 0; k0 < K; k0 += 4) {
    v2f a0 = *(const v2f*)(ap);
    v2f a1 = *(const v2f*)(ap + 16 * K);
    v2f b0 = *(const v2f*)(bp);
    v2f b1 = *(const v2f*)(bp + 16 * K);
    v2f b2 = *(const v2f*)(bp + 32 * K);
    v2f b3 = *(const v2f*)(bp + 48 * K);
    // 8 args: (neg_a, A, neg_b, B, c_mod, C, reuse_a, reuse_b)
    acc[0] = __builtin_amdgcn_wmma_f32_16x16x4_f32(false, a0, false, b0, (short)0, acc[0], false, false);
    acc[1] = __builtin_amdgcn_wmma_f32_16x16x4_f32(false, a0, false, b1, (short)0, acc[1], false, false);
    acc[2] = __builtin_amdgcn_wmma_f32_16x16x4_f32(false, a0, false, b2, (short)0, acc[2], false, false);
    acc[3] = __builtin_amdgcn_wmma_f32_16x16x4_f32(false, a0, false, b3, (short)0, acc[3], false, false);
    acc[4] = __builtin_amdgcn_wmma_f32_16x16x4_f32(false, a1, false, b0, (short)0, acc[4], false, false);
    acc[5] = __builtin_amdgcn_wmma_f32_16x16x4_f32(false, a1, false, b1, (short)0, acc[5], false, false);
    acc[6] = __builtin_amdgcn_wmma_f32_16x16x4_f32(false, a1, false, b2, (short)0, acc[6], false, false);
    acc[7] = __builtin_amdgcn_wmma_f32_16x16x4_f32(false, a1, false, b3, (short)0, acc[7], false, false);
    ap += 4;
    bp += 4;
  }

  #pragma unroll
  for (int t = 0; t < 4; ++t) {
    int col = n0 + t * 16 + laneN;
    if (col < N) {                       // only the store is guarded
      float bv = bias[col];
      float* cp = Cout + col;
      #pragma unroll
      for (int r = 0; r < 8; ++r) {
        int row = m0 + r + (hi << 3);
        cp[row * N]        = acc[t][r]     + bv;   // rows m0   .. m0+15
        cp[(row + 16) * N] = acc[4 + t][r] + bv;   // rows m0+16.. m0+31
      }
    }
  }
}

// -------------------------------------------------------------------------
// Depthwise 5x5 conv (SAME, zero pad) + LayerNorm(C) + exact GELU -> u
// Block = 192 threads = one pixel; wave32 butterfly + LDS for reductions.
// -------------------------------------------------------------------------
__global__ __launch_bounds__(192) void dwln_gelu(
    const float* __restrict__ x,  const float* __restrict__ dww,
    const float* __restrict__ dwb, const float* __restrict__ lng,
    const float* __restrict__ lnb, float* __restrict__ u)
{
  __shared__ float sred[8];
  int c   = threadIdx.x;
  int pix = blockIdx.x;
  int n = pix >> 12;
  int h = (pix >> 6) & 63;
  int w = pix & 63;

  float acc = dwb[c];
  #pragma unroll
  for (int i = 0; i < 5; ++i) {
    int y = h - 2 + i;
    if ((unsigned)y >= 64u) continue;
    #pragma unroll
    for (int j = 0; j < 5; ++j) {
      int xx = w - 2 + j;
      if ((unsigned)xx >= 64u) continue;
      acc += x[(((n * 64 + y) * 64) + xx) * C_ + c] * dww[(i * 5 + j) * C_ + c];
    }
  }

  int lane = c & 31, wid = c >> 5;
  float v = acc;
  #pragma unroll
  for (int o = 16; o; o >>= 1) v += __shfl_xor(v, o, 32);
  if (lane == 0) sred[wid] = v;
  __syncthreads();
  float mean = (sred[0] + sred[1] + sred[2] + sred[3] + sred[4] + sred[5]) * (1.0f / C_);
  float d = acc - mean;
  __syncthreads();
  v = d * d;
  #pragma unroll
  for (int o = 16; o; o >>= 1) v += __shfl_xor(v, o, 32);
  if (lane == 0) sred[wid] = v;
  __syncthreads();
  float var = (sred[0] + sred[1] + sred[2] + sred[3] + sred[4] + sred[5]) * (1.0f / C_);

  float t = d * rsqrtf(var + 1e-5f) * lng[c] + lnb[c];
  float g = 0.5f * t * (1.0f + erff(t * 0.70710678118654752f));  // exact GELU
  u[pix * C_ + c] = g;
}

// -------------------------------------------------------------------------
// In-place softmax over P=9 per (pixel, group). tid = pix*G + g, so the
// 9 scores live at m + tid*9 contiguously.
// -------------------------------------------------------------------------
__global__ __launch_bounds__(256) void mask_softmax(float* __restrict__ m, int total)
{
  int tid = blockIdx.x * blockDim.x + threadIdx.x;
  if (tid >= total) return;
  float* p = m + (size_t)tid * 9;
  float v[9], mx = -3.402823466e38f;
  #pragma unroll
  for (int i = 0; i < 9; ++i) { v[i] = p[i]; mx = fmaxf(mx, v[i]); }
  float s = 0.0f;
  #pragma unroll
  for (int i = 0; i < 9; ++i) { v[i] = expf(v[i] - mx); s += v[i]; }
  float inv = 1.0f / s;
  #pragma unroll
  for (int i = 0; i < 9; ++i) p[i] = v[i] * inv;
}

// -------------------------------------------------------------------------
// Deformable core. Analytically, the padded-grid-sample of the reference
// reduces to zero-padded bilinear at original-image coords:
//   sx = w + pts[p/3] + off_x,  sy = h + pts[p%3] + off_y,  pts = {-1,0,1}
// One thread per (pixel, group): 9 points x 4 corners x 16ch (float4 x4),
// then the sigmoid(cfs) gate blend with x_proj.
// -------------------------------------------------------------------------
__device__ __forceinline__ void corner_acc(float* acc, const float* __restrict__ xp,
                                           int n, int yi, int xi, int gbase, float wgt)
{
  if ((unsigned)xi >= 64u || (unsigned)yi >= 64u) return;
  const float4* p = (const float4*)(xp + (size_t)(((n * 64 + yi) * 64) + xi) * C_ + gbase);
  #pragma unroll
  for (int q = 0; q < 4; ++q) {
    float4 v = p[q];
    acc[q * 4 + 0] += wgt * v.x;
    acc[q * 4 + 1] += wgt * v.y;
    acc[q * 4 + 2] += wgt * v.z;
    acc[q * 4 + 3] += wgt * v.w;
  }
}

__global__ __launch_bounds__(256) void dcnv3_core(
    const float* __restrict__ xproj, const float* __restrict__ off,
    const float* __restrict__ msk,   const float* __restrict__ cfsr,
    float* __restrict__ y)
{
  int tid = blockIdx.x * 256 + threadIdx.x;
  int g   = tid % G_;
  int pix = tid / G_;
  if (pix >= L_) return;
  int n = pix >> 12;
  int h = (pix >> 6) & 63;
  int w = pix & 63;

  float acc[GC_];
  #pragma unroll
  for (int c = 0; c < GC_; ++c) acc[c] = 0.0f;

  const float* ob = off + (size_t)pix * GP2_ + g * (P_ * 2);
  const float* mb = msk + (size_t)pix * GP_  + g * P_;
  int gbase = g * GC_;

  #pragma unroll
  for (int p = 0; p < P_; ++p) {
    float ox = ob[p * 2 + 0];
    float oy = ob[p * 2 + 1];
    float mp = mb[p];
    float sx = (float)(w + (p / 3) - 1) + ox;   // x-offset varies along dim0 (torch order)
    float sy = (float)(h + (p % 3) - 1) + oy;
    float x0f = floorf(sx), y0f = floorf(sy);
    int   x0 = (int)x0f,    y0 = (int)y0f;
    float fx = sx - x0f,    fy = sy - y0f;
    corner_acc(acc, xproj, n, y0,     x0,     gbase, (1.0f - fx) * (1.0f - fy) * mp);
    corner_acc(acc, xproj, n, y0,     x0 + 1, gbase, fx * (1.0f - fy) * mp);
    corner_acc(acc, xproj, n, y0 + 1, x0,     gbase, (1.0f - fx) * fy * mp);
    corner_acc(acc, xproj, n, y0 + 1, x0 + 1, gbase, fx * fy * mp);
  }

  float cf = 1.0f / (1.0f + expf(-cfsr[pix * G_ + g]));
  const float* xb = xproj + (size_t)pix * C_ + gbase;
  float*       yb = y     + (size_t)pix * C_ + gbase;
  #pragma unroll
  for (int c = 0; c < GC_; ++c) yb[c] = acc[c] * (1.0f - cf) + xb[c] * cf;
}

// -------------------------------------------------------------------------
// Patch attention + final residual: one wave per pixel.
// Each lane owns 6 channels (lane + 32t). 9 neighbor dots -> butterfly
// reduce -> softmax -> unbiased std (ddof=1) -> out = x + x1*std.
// -------------------------------------------------------------------------
__global__ __launch_bounds__(256) void patch_attn_final(
    const float* __restrict__ x, const float* __restrict__ x1,
    float* __restrict__ out)
{
  int wid  = threadIdx.x >> 5;
  int lane = threadIdx.x & 31;
  int pix  = blockIdx.x * 8 + wid;
  int n = pix >> 12;
  int h = (pix >> 6) & 63;
  int w = pix & 63;

  const float* cb = x1 + (size_t)pix * C_;
  float cv[6];
  #pragma unroll
  for (int t = 0; t < 6; ++t) cv[t] = cb[lane + 32 * t];

  float s[9];
  #pragma unroll
  for (int k = 0; k < 9; ++k) {
    int yy = h + k / 3 - 1;
    int xx = w + k % 3 - 1;
    float part = 0.0f;
    if ((unsigned)yy < 64u && (unsigned)xx < 64u) {   // wave-uniform condition
      const float* nb = x1 + (size_t)(((n * 64 + yy) * 64) + xx) * C_;
      #pragma unroll
      for (int t = 0; t < 6; ++t) part += cv[t] * nb[lane + 32 * t];
    }
    #pragma unroll
    for (int o = 16; o; o >>= 1) part += __shfl_xor(part, o, 32);
    s[k] = part;
  }

  float mx = s[0];
  #pragma unroll
  for (int k = 1; k < 9; ++k) mx = fmaxf(mx, s[k]);
  float sum = 0.0f;
  #pragma unroll
  for (int k = 0; k < 9; ++k) { s[k] = expf(s[k] - mx); sum += s[k]; }
  float inv = 1.0f / sum;
  const float mean = 1.0f / 9.0f;   // softmax values sum to 1
  float var = 0.0f;
  #pragma unroll
  for (int k = 0; k < 9; ++k) { float d = s[k] * inv - mean; var += d * d; }
  float sd = sqrtf(var * (1.0f / 8.0f));   // ddof = 1

  const float* xb = x + (size_t)pix * C_;
  float*       ob = out + (size_t)pix * C_;
  #pragma unroll
  for (int t = 0; t < 6; ++t) {
    int ch = lane + 32 * t;
    ob[ch] = xb[ch] + cb[ch] * sd;
  }
}

// -------------------------------------------------------------------------
// Host orchestration. Workspace layout (floats), total ~94.6 MB:
//   [0)         x_proj  L*192
//   [6291456)   u       L*192   (reused as y after GEMMs on u are done)
//   [12582912)  offset  L*216   (reused as x1 after dcnv3 consumes it)
//   [19660800)  mask    L*108
//   [23199744)  cfs     L*12
//   [23592960)  Bt pack 256*192 (sequential reuse across the 5 GEMMs)
// -------------------------------------------------------------------------
extern "C" void kernel_launch(void* const* d_in, const int* in_sizes, int n_in,
                              void* d_out, int out_size, void* d_ws, size_t ws_size,
                              hipStream_t stream) {
  const float* x    = (const float*)d_in[0];
  const float* dww  = (const float*)d_in[1];
  const float* dwb  = (const float*)d_in[2];
  const float* lng  = (const float*)d_in[3];
  const float* lnb  = (const float*)d_in[4];
  const float* offw = (const float*)d_in[5];
  const float* offbias = (const float*)d_in[6];
  const float* mskw = (const float*)d_in[7];
  const float* mskbias = (const float*)d_in[8];
  const float* inw  = (const float*)d_in[9];
  const float* inbias  = (const float*)d_in[10];
  const float* outw = (const float*)d_in[11];
  const float* outbias = (const float*)d_in[12];
  const float* cfsw = (const float*)d_in[13];
  const float* cfsbias = (const float*)d_in[14];

  float* ws    = (float*)d_ws;
  float* xproj = ws;
  float* u     = ws + 6291456;
  float* offs  = ws + 12582912;
  float* maskb = ws + 19660800;
  float* cfsb  = ws + 23199744;
  float* btbuf = ws + 23592960;   // 256*192 floats
  float* y  = u;     // u dead after the three GEMMs on it
  float* x1 = offs;  // offsets dead after dcnv3_core

  auto gemm = [&](const float* A, const float* B, const float* bias, float* Cc, int N) {
    int Npad   = ((N + 63) / 64) * 64;
    int packTotal = Npad * C_;
    pack_bt<<<(packTotal + 255) / 256, 256, 0, stream>>>(B, btbuf, N, C_, packTotal);
    int tilesN4 = Npad / 64;
    int waves   = (L_ / 32) * tilesN4;           // 32-row tiles
    gemm_wmma_f32<<<(waves + 3) / 4, 128, 0, stream>>>(A, btbuf, bias, Cc, N, C_, tilesN4, waves);
  };

  gemm(x, inw, inbias, xproj, C_);                               // x_proj
  dwln_gelu<<<L_, 192, 0, stream>>>(x, dww, dwb, lng, lnb, u);   // u
  gemm(u, offw, offbias, offs, GP2_);                            // offsets (216)
  gemm(u, mskw, mskbias, maskb, GP_);                            // mask scores (108)
  gemm(u, cfsw, cfsbias, cfsb, G_);                              // cfs logits (12)
  mask_softmax<<<(L_ * G_) / 256, 256, 0, stream>>>(maskb, L_ * G_);
  dcnv3_core<<<(L_ * G_) / 256, 256, 0, stream>>>(xproj, offs, maskb, cfsb, y);
  gemm(y, outw, outbias, x1, C_);                                // x1 = out proj
  patch_attn_final<<<L_ / 8, 256, 0, stream>>>(x, x1, (float*)d_out);
}